// CodebookEncoder_26173530701857
// MI455X (gfx1250) — compile-verified
//
#include <hip/hip_runtime.h>
#include <stdint.h>

// ---------------------------------------------------------------------------
// CodebookEncoder for MI455X (gfx1250)
//
// Strategy (bandwidth-bound gather workload, ~18us HBM floor):
//  * Stage the whole 192KB codebook (6 LODs x 256 x 32 f32) into LDS once per
//    workgroup using the Tensor Data Mover (TENSOR_LOAD_TO_LDS), with TDM
//    pad insertion (16B after every 128B) so rows live at a 36-float stride
//    -> random per-lane row gathers spread over 16 LDS bank groups instead of 2.
//  * Lane-per-point: hash 8 corners, 4B gathers of the int64 index table's low
//    word (table is 12.6MB -> L2 resident), 8x ds_load_b128 per corner from
//    LDS, FMA into 32 accumulator VGPRs per LOD.
//  * Each lane's 32-float LOD result is one 128B-aligned cacheline of the
//    output -> 8 nontemporal b128 stores = full-line streaming writes.
// ---------------------------------------------------------------------------

typedef unsigned int uint32x4 __attribute__((ext_vector_type(4)));
typedef int          int32x8  __attribute__((ext_vector_type(8)));
typedef int          int32x4  __attribute__((ext_vector_type(4)));
typedef float        f32x4    __attribute__((ext_vector_type(4)));

#define DEPTH      6
#define FEAT       32
#define TABLE_SZ   262144           // 2^18 slots per LOD
#define TABLE_MASK 0x3FFFFu
#define ROWF       36               // padded row stride in floats (128B + 16B pad)
#define LODF       (256 * ROWF)     // floats per LOD in LDS (9216)
#define LDS_BYTES  (DEPTH * LODF * 4)   // 221184 B (<= 320KB/WGP)

__global__ __launch_bounds__(256, 1)
void CodebookEncoder_gfx1250_kernel(const float* __restrict__ coords,
                                    const float* __restrict__ codebook,
                                    const int*   __restrict__ itab,   // int64 table viewed as int pairs (LE)
                                    float*       __restrict__ out,
                                    int npts)
{
    extern __shared__ float lds[];

    // ---- Stage codebook -> LDS via Tensor Data Mover (wave 0 only) ----
    if (threadIdx.x < 32u) {
        uint64_t ga = (uint64_t)(uintptr_t)codebook;

        // D# group 0: count=1, lds_addr=0, 57-bit global addr, type=2 ("image")
        uint32x4 g0;
        g0.x = 1u;
        g0.y = 0u;
        g0.z = (unsigned)(ga & 0xFFFFFFFFu);
        g0.w = (unsigned)((ga >> 32) & 0x01FFFFFFu) | (2u << 30);

        // D# group 1:
        //  [15:0] workgroup_mask=0 (not in cluster), [17:16] data_size=2 (4B),
        //  [20] pad_enable=1, [24:22] pad_interval=4 (every 32 DWORDs),
        //  [31:25] pad_amount=3 (4 DWORDs = 16B)  => LDS row stride 144B
        int32x8 g1;
        g1[0] = (int)((2u << 16) | (1u << 20) | (4u << 22) | (3u << 25));
        g1[1] = (int)0xC0000000u;   // tensor_dim0[15:0]=49152 in bits[63:48]
        g1[2] = (int)0x00010000u;   // tensor_dim0[31:16]=0 ; tensor_dim1[15:0]=1
        g1[3] = (int)0xC0000000u;   // tensor_dim1[31:16]=0 ; tile_dim0=49152
        g1[4] = 1;                  // tile_dim1=1 ; tile_dim2=0
        g1[5] = 49152;              // tensor_dim0_stride[31:0]
        g1[6] = 0;
        g1[7] = 0;

        int32x4 g2 = (int32x4)0;    // dims 2/3 unused (1-D tensor)
        int32x4 g3 = (int32x4)0;
        int32x8 g4 = (int32x8)0;    // extra group (clang-23 6-arg form), unused

        __builtin_amdgcn_tensor_load_to_lds(g0, g1, g2, g3, g4, 0);
        __builtin_amdgcn_s_wait_tensorcnt(0);
    }
    __syncthreads();   // release the other 7 waves once LDS is populated

    const unsigned P1 = 2654435761u, P2 = 805459861u;

    for (int n = blockIdx.x * blockDim.x + threadIdx.x; n < npts;
         n += gridDim.x * blockDim.x)
    {
        const float cx = coords[3 * n + 0];
        const float cy = coords[3 * n + 1];
        const float cz = coords[3 * n + 2];

        #pragma unroll
        for (int l = 0; l < DEPTH; ++l) {
            const float res = (float)(2 << l);            // 2^(l+1)
            const float fx = cx * res, fy = cy * res, fz = cz * res;
            float xi = fminf(fmaxf(floorf(fx), 0.f), res - 1.f);
            float yi = fminf(fmaxf(floorf(fy), 0.f), res - 1.f);
            float zi = fminf(fmaxf(floorf(fz), 0.f), res - 1.f);
            const float tx = fx - xi, ty = fy - yi, tz = fz - zi;
            const float sx = 1.f - tx, sy = 1.f - ty, sz = 1.f - tz;
            const unsigned ux = (unsigned)xi, uy = (unsigned)yi, uz = (unsigned)zi;
            const unsigned salt = (unsigned)l * 2166136261u;

            f32x4 acc[FEAT / 4];
            #pragma unroll
            for (int j = 0; j < FEAT / 4; ++j) acc[j] = (f32x4)0.0f;

            const f32x4* __restrict__ lcb = (const f32x4*)(lds + l * LODF);

            #pragma unroll
            for (int k = 0; k < 8; ++k) {
                const unsigned lx = ux + ((k >> 2) & 1);
                const unsigned ly = uy + ((k >> 1) & 1);
                const unsigned lz = uz + (k & 1);
                const unsigned h  = (lx ^ (ly * P1) ^ (lz * P2)) + salt;
                const unsigned slot = h & TABLE_MASK;
                // int64 table: low 32 bits hold the value (0..255)
                const unsigned idx =
                    (unsigned)itab[2 * ((size_t)l * TABLE_SZ + slot)] & 0xFFu;

                const float w = ((k & 4) ? tx : sx) *
                                ((k & 2) ? ty : sy) *
                                ((k & 1) ? tz : sz);

                const f32x4* __restrict__ row = lcb + (size_t)idx * (ROWF / 4);
                #pragma unroll
                for (int j = 0; j < FEAT / 4; ++j) {
                    f32x4 c = row[j];          // ds_load_b128, padded stride
                    acc[j] += w * c;
                }
            }

            // 32 floats = one 128B-aligned cacheline of the output
            f32x4* __restrict__ op =
                (f32x4*)(out + (size_t)n * (DEPTH * FEAT) + l * FEAT);
            #pragma unroll
            for (int j = 0; j < FEAT / 4; ++j)
                __builtin_nontemporal_store(acc[j], op + j);
        }
    }
}

extern "C" void kernel_launch(void* const* d_in, const int* in_sizes, int n_in,
                              void* d_out, int out_size, void* d_ws, size_t ws_size,
                              hipStream_t stream) {
    const float* coords   = (const float*)d_in[0];
    const float* codebook = (const float*)d_in[1];
    const int*   itab     = (const int*)d_in[2];   // int64 viewed as int pairs
    float*       out      = (float*)d_out;
    const int    npts     = in_sizes[0] / 3;       // 524288

    // 216KB dynamic LDS > default cap; attribute set is idempotent & capture-safe.
    (void)hipFuncSetAttribute((const void*)CodebookEncoder_gfx1250_kernel,
                              hipFuncAttributeMaxDynamicSharedMemorySize,
                              LDS_BYTES);

    dim3 grid(1024), block(256);
    CodebookEncoder_gfx1250_kernel<<<grid, block, LDS_BYTES, stream>>>(
        coords, codebook, itab, out, npts);
}